// SubComplexIncidenceConv_6227702579781
// MI455X (gfx1250) — compile-verified
//
#include <hip/hip_runtime.h>

typedef float v2f __attribute__((ext_vector_type(2)));
typedef float v8f __attribute__((ext_vector_type(8)));

#define N_NODES 100000
#define DCH     128
#define BN_EPS  1e-5f
#define SZ      ((size_t)N_NODES * DCH)   // floats per feature buffer

// ---------------------------------------------------------------------------
// K1: h0 = (1+eps)*x   (float4 vectorized), and zero the 1024-float stats area
// ---------------------------------------------------------------------------
__global__ __launch_bounds__(256)
void gin_init(const float* __restrict__ x, const float* __restrict__ eps,
              float* __restrict__ h0, float* __restrict__ stats)
{
    const long i = (long)blockIdx.x * blockDim.x + threadIdx.x;   // float4 index
    if (i < 1024) stats[i] = 0.0f;
    const float e = 1.0f + eps[0];
    float4 v = ((const float4*)x)[i];
    v.x *= e; v.y *= e; v.z *= e; v.w *= e;
    ((float4*)h0)[i] = v;
}

// ---------------------------------------------------------------------------
// K2: scatter-add. One wave per edge: 32 lanes x float4 = 128 channels.
// x and agg are L2-resident (51.2 MB each vs 192 MB L2) -> L2 atomic bound.
// ---------------------------------------------------------------------------
__global__ __launch_bounds__(256)
void gin_scatter(const float* __restrict__ x, const int* __restrict__ src,
                 const int* __restrict__ dst, float* __restrict__ agg, int nEdges)
{
    const int e = blockIdx.x * 8 + (threadIdx.x >> 5);
    if (e >= nEdges) return;
    const int lane = threadIdx.x & 31;
    const int s = src[e];
    const int d = dst[e];
    const float4 v = ((const float4*)(x + (size_t)s * DCH))[lane];
    float* p = agg + (size_t)d * DCH + lane * 4;
    atomicAdd(p + 0, v.x);
    atomicAdd(p + 1, v.y);
    atomicAdd(p + 2, v.z);
    atomicAdd(p + 3, v.w);
}

// ---------------------------------------------------------------------------
// K3/K5: WMMA f32 GEMM:  out[N,128] = f(A)[N,128] @ W[128,128] + bias
// 8 waves per block; wave w computes the 16x16 tile at (mtile, ntile=w).
// K-loop: 32 steps of V_WMMA_F32_16X16X4_F32.
// If APPLY_BN: A element -> relu(a*in_scale[k] + in_shift[k]) (fused BN1+ReLU).
// Epilogue accumulates per-channel sum / sum-of-squares for the next BN.
// ---------------------------------------------------------------------------
template <int APPLY_BN>
__global__ __launch_bounds__(256)
void gin_gemm_wmma(const float* __restrict__ A, const float* __restrict__ W,
                   const float* __restrict__ bias,
                   const float* __restrict__ in_scale, const float* __restrict__ in_shift,
                   float* __restrict__ out,
                   float* __restrict__ col_sum, float* __restrict__ col_ssq)
{
    const int lane  = threadIdx.x & 31;
    const int wave  = threadIdx.x >> 5;        // ntile 0..7
    const int mtile = blockIdx.x;              // 0..6249 (100000/16)
    const int mrow  = lane & 15;               // A row within tile (lane layout)
    const int khalf = (lane >> 4) * 2;         // lanes 0-15 -> K0/K1, 16-31 -> K2/K3
    const int coln  = wave * 16 + mrow;        // B column this lane supplies

    const float* Arow = A + (size_t)(mtile * 16 + mrow) * DCH;

    v8f c = {0.f, 0.f, 0.f, 0.f, 0.f, 0.f, 0.f, 0.f};
    for (int kb = 0; kb < DCH; kb += 4) {
        const int k0 = kb + khalf;
        v2f a, b;
        a.x = Arow[k0];
        a.y = Arow[k0 + 1];
        if (APPLY_BN) {
            a.x = fmaxf(a.x * in_scale[k0]     + in_shift[k0],     0.0f);
            a.y = fmaxf(a.y * in_scale[k0 + 1] + in_shift[k0 + 1], 0.0f);
        }
        b.x = W[(size_t)k0 * DCH + coln];
        b.y = W[(size_t)(k0 + 1) * DCH + coln];
        // (neg_a, A, neg_b, B, c_mod, C, reuse_a, reuse_b)
        c = __builtin_amdgcn_wmma_f32_16x16x4_f32(false, a, false, b,
                                                  (short)0, c, false, false);
    }

    // Epilogue: C/D layout -> VGPR i holds rows {i, i+8} split across lane halves.
    const int col   = wave * 16 + (lane & 15);
    const int rbase = mtile * 16 + ((lane >> 4) << 3);
    const float bv  = bias[col];
    float ps = 0.0f, pq = 0.0f;
#pragma unroll
    for (int i = 0; i < 8; ++i) {
        const float v = c[i] + bv;
        out[(size_t)(rbase + i) * DCH + col] = v;
        ps += v;
        pq += v * v;
    }
    atomicAdd(&col_sum[col], ps);   // partial over 8 rows of this tile
    atomicAdd(&col_ssq[col], pq);
}

// ---------------------------------------------------------------------------
// K4/K6: fold batch stats into affine scale/shift:
//   mu = S/N; var = Q/N - mu^2; scale = g*rsqrt(var+eps); shift = be - mu*scale
// ---------------------------------------------------------------------------
__global__ void gin_bn_finalize(const float* __restrict__ s, const float* __restrict__ q,
                                const float* __restrict__ g, const float* __restrict__ be,
                                float* __restrict__ scale, float* __restrict__ shift)
{
    const int c = threadIdx.x;
    const float inv_n = 1.0f / (float)N_NODES;
    const float mu  = s[c] * inv_n;
    const float var = q[c] * inv_n - mu * mu;
    const float sc  = g[c] * __frsqrt_rn(var + BN_EPS);
    scale[c] = sc;
    shift[c] = be[c] - mu * sc;
}

// ---------------------------------------------------------------------------
// K7: out = relu(t2*scale2[col] + shift2[col]) (float4 vectorized)
// ---------------------------------------------------------------------------
__global__ __launch_bounds__(256)
void gin_bnrelu_out(const float* __restrict__ t2, const float* __restrict__ scale,
                    const float* __restrict__ shift, float* __restrict__ out)
{
    const long i = (long)blockIdx.x * blockDim.x + threadIdx.x;   // float4 index
    const int cb = (int)((i * 4) & (DCH - 1));
    float4 v = ((const float4*)t2)[i];
    v.x = fmaxf(v.x * scale[cb + 0] + shift[cb + 0], 0.0f);
    v.y = fmaxf(v.y * scale[cb + 1] + shift[cb + 1], 0.0f);
    v.z = fmaxf(v.z * scale[cb + 2] + shift[cb + 2], 0.0f);
    v.w = fmaxf(v.w * scale[cb + 3] + shift[cb + 3], 0.0f);
    ((float4*)out)[i] = v;
}

// ---------------------------------------------------------------------------
extern "C" void kernel_launch(void* const* d_in, const int* in_sizes, int n_in,
                              void* d_out, int out_size, void* d_ws, size_t ws_size,
                              hipStream_t stream)
{
    const float* x   = (const float*)d_in[0];
    const int*   ei  = (const int*)  d_in[1];
    const float* W1  = (const float*)d_in[2];
    const float* b1  = (const float*)d_in[3];
    const float* g1  = (const float*)d_in[4];
    const float* be1 = (const float*)d_in[5];
    const float* W2  = (const float*)d_in[6];
    const float* b2  = (const float*)d_in[7];
    const float* g2  = (const float*)d_in[8];
    const float* be2 = (const float*)d_in[9];
    const float* eps = (const float*)d_in[10];
    float* out = (float*)d_out;

    const int nEdges = in_sizes[1] / 2;          // edge_index is [2, E]
    const int* src = ei;
    const int* dst = ei + nEdges;

    // Workspace layout
    float* bufA  = (float*)d_ws;                 // h0, later overwritten by t2
    float* bufB  = bufA + SZ;                    // t1
    float* stats = bufB + SZ;                    // 1024 floats
    float* sum1   = stats + 0;
    float* ssq1   = stats + 128;
    float* scale1 = stats + 256;
    float* shift1 = stats + 384;
    float* sum2   = stats + 512;
    float* ssq2   = stats + 640;
    float* scale2 = stats + 768;
    float* shift2 = stats + 896;

    const int vec4   = (int)(SZ / 4);            // 3,200,000 float4 elements
    const int vblk   = vec4 / 256;               // 12,500 blocks
    const int mblk   = N_NODES / 16;             // 6,250 blocks
    const int eblk   = (nEdges + 7) / 8;         // 8 edges (waves) per block

    // K1: h0 = (1+eps)*x, zero stats
    gin_init<<<vblk, 256, 0, stream>>>(x, eps, bufA, stats);
    // K2: agg += x[src] scattered onto dst (L2-resident f32 atomics)
    gin_scatter<<<eblk, 256, 0, stream>>>(x, src, dst, bufA, nEdges);
    // K3: t1 = h0 @ W1 + b1, accumulate BN1 stats
    gin_gemm_wmma<0><<<mblk, 256, 0, stream>>>(bufA, W1, b1, nullptr, nullptr,
                                               bufB, sum1, ssq1);
    // K4: BN1 scale/shift
    gin_bn_finalize<<<1, 128, 0, stream>>>(sum1, ssq1, g1, be1, scale1, shift1);
    // K5: t2 = relu(BN1(t1)) @ W2 + b2, accumulate BN2 stats
    gin_gemm_wmma<1><<<mblk, 256, 0, stream>>>(bufB, W2, b2, scale1, shift1,
                                               bufA, sum2, ssq2);
    // K6: BN2 scale/shift
    gin_bn_finalize<<<1, 128, 0, stream>>>(sum2, ssq2, g2, be2, scale2, shift2);
    // K7: out = relu(BN2(t2))
    gin_bnrelu_out<<<vblk, 256, 0, stream>>>(bufA, scale2, shift2, out);
}